// YoloLoss_50981261804097
// MI455X (gfx1250) — compile-verified
//
#include <hip/hip_runtime.h>
#include <hip/hip_bf16.h>
#include <math.h>

// predictions: (N,7,7,90) fp32 ; target: (N,7,7,85) fp32
// outputs: (coords_loss, obj_loss, noobj_loss, classes_loss) -> d_out[0..3] fp32

typedef float v2f __attribute__((ext_vector_type(2)));
typedef float v8f __attribute__((ext_vector_type(8)));

__device__ __forceinline__ float iou_cxcywh(float a0, float a1, float a2, float a3,
                                            float b0, float b1, float b2, float b3) {
    float ax1 = a0 - a2 * 0.5f, ay1 = a1 - a3 * 0.5f;
    float ax2 = a0 + a2 * 0.5f, ay2 = a1 + a3 * 0.5f;
    float bx1 = b0 - b2 * 0.5f, by1 = b1 - b3 * 0.5f;
    float bx2 = b0 + b2 * 0.5f, by2 = b1 + b3 * 0.5f;
    float iw = fmaxf(fminf(ax2, bx2) - fmaxf(ax1, bx1), 0.f);
    float ih = fmaxf(fminf(ay2, by2) - fmaxf(ay1, by1), 0.f);
    float inter = iw * ih;
    float area_a = (ax2 - ax1) * (ay2 - ay1);
    float area_b = (bx2 - bx1) * (by2 - by1);
    return inter / (area_a + area_b - inter);
}

// Block (256 thr = 8 wave32) reduction of 4 accumulators -> atomicAdd into out[0..3].
// Final 32->4 stage uses two chained V_WMMA_F32_16X16X4_F32 with B = ones:
// D[m][n] = sum_k A[m][k], so rows 0..3 of D hold the four block totals (fp32-exact).
__device__ __forceinline__ void block_reduce4_atomic(float v0, float v1, float v2, float v3,
                                                     float* out) {
    #pragma unroll
    for (int off = 16; off; off >>= 1) {
        v0 += __shfl_down(v0, off, 32);
        v1 += __shfl_down(v1, off, 32);
        v2 += __shfl_down(v2, off, 32);
        v3 += __shfl_down(v3, off, 32);
    }
    __shared__ float parts[8][4];
    const int lane = threadIdx.x & 31;
    const int wave = threadIdx.x >> 5;
    if (lane == 0) {
        parts[wave][0] = v0; parts[wave][1] = v1;
        parts[wave][2] = v2; parts[wave][3] = v3;
    }
    __syncthreads();
    if (wave == 0) {
        const int m  = lane & 15;
        const int hi = (lane >= 16) ? 1 : 0;
#if __has_builtin(__builtin_amdgcn_wmma_f32_16x16x4_f32)
        // fp32 A-matrix 16x4 layout: lanes 0-15 hold K0 (vgpr0)/K1 (vgpr1),
        // lanes 16-31 hold K2/K3; M = lane%16. Row j sums wave partials of acc j.
        v2f a1 = (v2f){0.f, 0.f}, a2 = (v2f){0.f, 0.f};
        if (m < 4) {
            a1[0] = parts[hi ? 2 : 0][m];
            a1[1] = parts[hi ? 3 : 1][m];
            a2[0] = parts[hi ? 6 : 4][m];
            a2[1] = parts[hi ? 7 : 5][m];
        }
        v2f b = (v2f){1.f, 1.f};              // B layout-independent: all ones
        v8f c = (v8f){0.f, 0.f, 0.f, 0.f, 0.f, 0.f, 0.f, 0.f};
        c = __builtin_amdgcn_wmma_f32_16x16x4_f32(false, a1, false, b, (short)0, c, false, false);
        c = __builtin_amdgcn_wmma_f32_16x16x4_f32(false, a2, false, b, (short)0, c, false, false);
        // D vgpr r = row M=r on lanes 0-15; lane 0 reads rows 0..3.
        if (lane == 0) {
            atomicAdd(out + 0, c[0]);
            atomicAdd(out + 1, c[1]);
            atomicAdd(out + 2, c[2]);
            atomicAdd(out + 3, c[3]);
        }
#else
        (void)hi;
        if (m < 4 && lane < 4) {
            float s = 0.f;
            #pragma unroll
            for (int w = 0; w < 8; ++w) s += parts[w][lane];
            atomicAdd(out + lane, s);
        }
#endif
    }
}

__global__ void zero4_kernel(float* out) {
    if (threadIdx.x < 4) out[threadIdx.x] = 0.f;
}

// Box / obj / noobj losses: one thread per cell (15 streamed floats per cell).
__global__ __launch_bounds__(256) void box_kernel(const float* __restrict__ preds,
                                                  const float* __restrict__ tgt,
                                                  float* __restrict__ out, int ncells) {
    float coords = 0.f, objl = 0.f, noobjl = 0.f;
    for (int cell = blockIdx.x * 256 + threadIdx.x; cell < ncells; cell += gridDim.x * 256) {
        const float* P = preds + (size_t)cell * 90;
        const float* T = tgt   + (size_t)cell * 85;
        float obj = T[80];
        float tb0 = T[81], tb1 = T[82], tb2 = T[83], tb3 = T[84];
        float q10 = __builtin_nontemporal_load(P + 86);
        float q11 = __builtin_nontemporal_load(P + 87);
        float q12 = __builtin_nontemporal_load(P + 88);
        float q13 = __builtin_nontemporal_load(P + 89);
        float q20 = __builtin_nontemporal_load(P + 81);
        float q21 = __builtin_nontemporal_load(P + 82);
        float q22 = __builtin_nontemporal_load(P + 83);
        float q23 = __builtin_nontemporal_load(P + 84);
        float conf1 = __builtin_nontemporal_load(P + 85);
        float conf2 = __builtin_nontemporal_load(P + 80);

        float iou1 = iou_cxcywh(q10, q11, q12, q13, tb0, tb1, tb2, tb3);
        float iou2 = iou_cxcywh(q20, q21, q22, q23, tb0, tb1, tb2, tb3);
        // numpy/jax argmax NaN semantics: first NaN wins, ties -> index 0
        bool pick1 = __builtin_isnan(iou1) || (!__builtin_isnan(iou2) && (iou1 >= iou2));
        float m1 = pick1 ? 1.f : 0.f;
        float m2 = 1.f - m1;

        float pb0 = obj * (m1 * q10 + m2 * q20);           // cx (only comp in center_loss)
        float pbw = obj * (m1 * q12 + m2 * q22);
        float pbh = obj * (m1 * q13 + m2 * q23);

        float dc = pb0 - tb0;
        float sw = (pbw > 0.f) ? 1.f : ((pbw < 0.f) ? -1.f : 0.f);
        float sh = (pbh > 0.f) ? 1.f : ((pbh < 0.f) ? -1.f : 0.f);
        float dw = sw * sqrtf(fabsf(pbw) + 1e-6f) - sqrtf(tb2);
        float dh = sh * sqrtf(fabsf(pbh) + 1e-6f) - sqrtf(tb3);
        coords += 5.f * (dc * dc) + 5.f * (dw * dw + dh * dh);

        float op = conf1 * m1 + conf2 * m2;
        float e1 = obj * op - obj;
        objl += e1 * e1;
        float e2 = (1.f - obj) * op - obj;
        noobjl += 0.5f * (e2 * e2);
    }
    block_reduce4_atomic(coords, objl, noobjl, 0.f, out);
}

// Classes loss: element-parallel over (cell, class) for coalesced streaming (93% of bytes).
__global__ __launch_bounds__(256) void classes_kernel(const float* __restrict__ preds,
                                                      const float* __restrict__ tgt,
                                                      float* __restrict__ out, int ncells) {
    const int total = ncells * 80;
    float acc = 0.f;
    for (int i = blockIdx.x * 256 + threadIdx.x; i < total; i += gridDim.x * 256) {
        int cell = i / 80;
        int c    = i - cell * 80;
        float p   = __builtin_nontemporal_load(preds + (size_t)cell * 90 + c);
        float t   = __builtin_nontemporal_load(tgt   + (size_t)cell * 85 + c);
        float obj = tgt[(size_t)cell * 85 + 80];   // shared by 80 threads -> keep cacheable
        float d = p * obj - t;
        acc += d * d;
    }
    block_reduce4_atomic(0.f, 0.f, 0.f, acc, out);
}

extern "C" void kernel_launch(void* const* d_in, const int* in_sizes, int n_in,
                              void* d_out, int out_size, void* d_ws, size_t ws_size,
                              hipStream_t stream) {
    const float* preds = (const float*)d_in[0];
    const float* tgt   = (const float*)d_in[1];
    float* out = (float*)d_out;
    const int ncells = in_sizes[0] / 90;   // N*S*S

    zero4_kernel<<<1, 32, 0, stream>>>(out);

    int boxBlocks = (ncells + 255) / 256;
    if (boxBlocks < 1) boxBlocks = 1;
    box_kernel<<<boxBlocks, 256, 0, stream>>>(preds, tgt, out, ncells);

    long long totalCls = (long long)ncells * 80;
    int clsBlocks = (int)((totalCls + 255) / 256);
    if (clsBlocks > 16384) clsBlocks = 16384;
    if (clsBlocks < 1) clsBlocks = 1;
    classes_kernel<<<clsBlocks, 256, 0, stream>>>(preds, tgt, out, ncells);
}